// SGATConv_2207613190378
// MI455X (gfx1250) — compile-verified
//
#include <hip/hip_runtime.h>

typedef float v2f __attribute__((ext_vector_type(2)));
typedef float v8f __attribute__((ext_vector_type(8)));

constexpr int NN  = 50000;   // nodes
constexpr int NE  = 800000;  // edges
constexpr int INF = 128;     // IN_FEATS
constexpr int EF  = 64;      // EDGE_FEATS
constexpr int HD  = 128;     // HEADS*OUT_FEATS
constexpr int NH  = 4;       // HEADS
constexpr float NEG_SLOPE = 0.2f;

// LDS row strides (floats): stride % 64 == 4 -> conflict-free ds_load_b64
// across all 32 lanes; stride*4 % 16 == 0 -> float4 stores stay aligned.
constexpr int LDA_NODE = 132;   // for 16x128 node tile (8448 B)
constexpr int LDA_EDGE = 68;    // for 16x64  edge tile (4352 B)

// ---- workspace layout (float offsets) ----
constexpr size_t WS_FEATSRC = 0;                             // NN*HD  (25.6 MB)
constexpr size_t WS_EBUF    = WS_FEATSRC + (size_t)NN * HD;  // NE*NH  (12.8 MB)
constexpr size_t WS_EL      = WS_EBUF    + (size_t)NE * NH;  // NN*NH
constexpr size_t WS_ER      = WS_EL      + (size_t)NN * NH;  // NN*NH
constexpr size_t WS_EMAX    = WS_ER      + (size_t)NN * NH;  // NN*NH
constexpr size_t WS_DENOM   = WS_EMAX    + (size_t)NN * NH;  // NN*NH
constexpr size_t WS_WEL     = WS_DENOM   + (size_t)NN * NH;  // 512
constexpr size_t WS_WER     = WS_WEL + 512;                  // 512
constexpr size_t WS_WEE     = WS_WER + 512;                  // 256

// ---------------- WMMA helper: D = A(16x4,f32) * B(4x16,f32) + C ----------------
__device__ inline v8f wmma_f32(v2f a, v2f b, v8f c) {
    // (neg_a, A, neg_b, B, c_mod, C, reuse_a, reuse_b)
    return __builtin_amdgcn_wmma_f32_16x16x4_f32(false, a, false, b, (short)0, c,
                                                 false, false);
}

// exact float atomic max via sign-split integer atomics
__device__ inline void atomicMaxF(float* addr, float v) {
    if (v >= 0.0f) atomicMax(reinterpret_cast<int*>(addr), __float_as_int(v));
    else           atomicMin(reinterpret_cast<unsigned int*>(addr), __float_as_uint(v));
}

// ---------------- K0: fold attn vectors into weights ----------------
// w_el[h][k] = sum_d attn_l[h,d] * W_fc[h*32+d][k]   (4x128)
// w_er likewise with attn_r; w_ee[h][k] with W_edge  (4x64)
__global__ __launch_bounds__(512) void build_attn_w(
    const float* __restrict__ Wfc, const float* __restrict__ Wedge,
    const float* __restrict__ attn_l, const float* __restrict__ attn_r,
    const float* __restrict__ attn_e,
    float* __restrict__ wel, float* __restrict__ wer, float* __restrict__ wee)
{
    const int t = threadIdx.x;              // 0..511
    {
        const int h = t >> 7, k = t & 127;
        float al = 0.f, ar = 0.f;
        #pragma unroll 4
        for (int d = 0; d < 32; ++d) {
            const float wf = Wfc[(size_t)(h * 32 + d) * INF + k];
            al += attn_l[h * 32 + d] * wf;
            ar += attn_r[h * 32 + d] * wf;
        }
        wel[t] = al; wer[t] = ar;
    }
    if (t < 256) {
        const int h = t >> 6, k = t & 63;
        float ae = 0.f;
        #pragma unroll 4
        for (int d = 0; d < 32; ++d)
            ae += attn_e[h * 32 + d] * Wedge[(size_t)(h * 32 + d) * EF + k];
        wee[t] = ae;
    }
}

// ---------------- K_init: emax = -inf, denom = 0 ----------------
__global__ void init_redux(float* __restrict__ emax, float* __restrict__ denom) {
    const int i = blockIdx.x * blockDim.x + threadIdx.x;
    if (i < NN * NH) {
        emax[i]  = __int_as_float(0xff800000); // -inf
        denom[i] = 0.0f;
    }
}

// ---------------- K1: node projection (WMMA f32, LDS-staged A) ----------------
// grid = (NN/16, 2); block = 256 (8 waves = 8 col-tiles of 16)
// y==0: feat_src = feat @ W_fc^T ;  y==1: out = feat @ W_res^T + bias
__global__ __launch_bounds__(256) void node_proj_wmma(
    const float* __restrict__ feat, const float* __restrict__ Wfc,
    const float* __restrict__ Wres, const float* __restrict__ bias,
    float* __restrict__ feat_src, float* __restrict__ out)
{
    __shared__ float sA[16 * LDA_NODE];

    const int t = threadIdx.x;
    // cooperative fill: 16x128 tile, 8 floats/thread, padded row stride
    {
        const int r = t >> 4, c = (t & 15) * 8;
        const float* g = feat + (size_t)(blockIdx.x * 16 + r) * INF + c;
        *(float4*)(sA + r * LDA_NODE + c)     = *(const float4*)(g);
        *(float4*)(sA + r * LDA_NODE + c + 4) = *(const float4*)(g + 4);
    }
    __syncthreads();

    const int lane = t & 31;
    const int wv   = t >> 5;               // col tile 0..7
    const int half = lane >> 4;            // K-pair select
    const int l16  = lane & 15;
    const int colg = wv * 16 + l16;        // B: N = lane%16
    const int kb   = half * 2;             // A/B: K base per half-wave
    const bool res = (blockIdx.y != 0);
    const float* __restrict__ W = res ? Wres : Wfc;

    v8f acc = {0.f, 0.f, 0.f, 0.f, 0.f, 0.f, 0.f, 0.f};
    const float* __restrict__ aL   = sA + l16 * LDA_NODE + kb;       // LDS
    const float* __restrict__ brow = W  + (size_t)colg * INF + kb;   // B[k][n] = W[n][k]
    #pragma unroll
    for (int k = 0; k < INF; k += 4) {
        v2f a = *(const v2f*)(aL + k);
        v2f b = *(const v2f*)(brow + k);
        acc = wmma_f32(a, b, acc);
    }
    // D layout: VGPR r -> M = r + 8*half, N = lane%16
    const int mbase = blockIdx.x * 16 + half * 8;
    if (res) {
        const float bv = bias[colg];
        #pragma unroll
        for (int r = 0; r < 8; ++r)
            out[(size_t)(mbase + r) * HD + colg] = acc[r] + bv;
    } else {
        #pragma unroll
        for (int r = 0; r < 8; ++r)
            feat_src[(size_t)(mbase + r) * HD + colg] = acc[r];
    }
}

// ---------------- K2: per-node attention scalars ----------------
__global__ __launch_bounds__(256) void node_attn(
    const float* __restrict__ feat,
    const float* __restrict__ wel, const float* __restrict__ wer,
    float* __restrict__ el, float* __restrict__ er)
{
    const int idx = blockIdx.x * blockDim.x + threadIdx.x;
    if (idx >= NN * NH) return;
    const int n = idx >> 2, h = idx & 3;
    const float* __restrict__ f  = feat + (size_t)n * INF;
    const float* __restrict__ wl = wel + h * INF;
    const float* __restrict__ wr = wer + h * INF;
    float al = 0.f, ar = 0.f;
    #pragma unroll 8
    for (int k = 0; k < INF; k += 4) {
        const float4 a = *(const float4*)(f + k);
        const float4 l = *(const float4*)(wl + k);
        const float4 r = *(const float4*)(wr + k);
        al += a.x * l.x + a.y * l.y + a.z * l.z + a.w * l.w;
        ar += a.x * r.x + a.y * r.y + a.z * r.z + a.w * r.w;
    }
    el[idx] = al; er[idx] = ar;
}

// ---------------- K3: edge logits + LeakyReLU + segment max ----------------
__global__ __launch_bounds__(256) void edge_logits(
    const float* __restrict__ edge_feat,
    const int* __restrict__ src, const int* __restrict__ dst,
    const float* __restrict__ wee,
    const float* __restrict__ el, const float* __restrict__ er,
    float* __restrict__ ebuf, float* __restrict__ emax)
{
    const int idx = blockIdx.x * blockDim.x + threadIdx.x;
    if (idx >= NE * NH) return;
    const int e = idx >> 2, h = idx & 3;
    const float* __restrict__ ef = edge_feat + (size_t)e * EF;
    const float* __restrict__ w  = wee + h * EF;
    float ee = 0.f;
    #pragma unroll 4
    for (int k = 0; k < EF; k += 4) {
        const float4 a = *(const float4*)(ef + k);
        const float4 b = *(const float4*)(w + k);
        ee += a.x * b.x + a.y * b.y + a.z * b.z + a.w * b.w;
    }
    const int s = src[e], d = dst[e];
    float x = el[s * NH + h] + er[d * NH + h] + ee;
    x = (x > 0.f) ? x : NEG_SLOPE * x;
    ebuf[idx] = x;
    atomicMaxF(&emax[d * NH + h], x);
}

// ---------------- K4: numerator exp + segment sum ----------------
__global__ __launch_bounds__(256) void edge_softmax_num(
    const int* __restrict__ dst, const float* __restrict__ emax,
    float* __restrict__ ebuf, float* __restrict__ denom)
{
    const int idx = blockIdx.x * blockDim.x + threadIdx.x;
    if (idx >= NE * NH) return;
    const int e = idx >> 2, h = idx & 3;
    const int d = dst[e];
    const float ex = __expf(ebuf[idx] - emax[d * NH + h]);
    ebuf[idx] = ex;
    unsafeAtomicAdd(&denom[d * NH + h], ex);
}

// ---------------- K5: fused edge projection (WMMA) + scatter-add ----------------
// grid = NE/16; block = 256 (8 waves = 8 col-tiles). Recomputes the edge
// projection tile in-register (avoids 410 MB feat_edge spill). A tile and all
// per-edge scalars (src/dst/softmax coeff) are staged in LDS once per block;
// epilogue per lane = L2-resident feat_src gather + global_atomic_add_f32.
__global__ __launch_bounds__(256) void edge_agg_wmma(
    const float* __restrict__ edge_feat, const float* __restrict__ Wedge,
    const int* __restrict__ src, const int* __restrict__ dst,
    const float* __restrict__ feat_src, const float* __restrict__ exbuf,
    const float* __restrict__ denom, float* __restrict__ out)
{
    __shared__ float sA[16 * LDA_EDGE];   // edge_feat tile
    __shared__ float sa[16][NH];          // attention coefficient a[e][h]
    __shared__ int   ssrc[16];
    __shared__ int   sdst[16];

    const int t = threadIdx.x;
    const int ebase = blockIdx.x * 16;

    // cooperative fill: 16x64 tile, 4 floats/thread, padded row stride
    {
        const int r = t >> 4, c = (t & 15) * 4;
        *(float4*)(sA + r * LDA_EDGE + c) =
            *(const float4*)(edge_feat + (size_t)(ebase + r) * EF + c);
    }
    if (t < 16) {
        ssrc[t] = src[ebase + t];
        sdst[t] = dst[ebase + t];
    }
    if (t < 64) {                         // a = exp(e - emax)/denom per (edge, head)
        const int le = t >> 2, h = t & 3;
        const int e  = ebase + le;
        const int d  = dst[e];
        sa[le][h] = exbuf[(size_t)e * NH + h] / denom[(size_t)d * NH + h];
    }
    __syncthreads();

    const int lane = t & 31;
    const int wv   = t >> 5;
    const int half = lane >> 4;
    const int l16  = lane & 15;
    const int colg = wv * 16 + l16;
    const int h    = colg >> 5;           // head for this column tile
    const int kb   = half * 2;

    v8f acc = {0.f, 0.f, 0.f, 0.f, 0.f, 0.f, 0.f, 0.f};
    const float* __restrict__ aL   = sA + l16 * LDA_EDGE + kb;      // LDS
    const float* __restrict__ brow = Wedge + (size_t)colg * EF + kb;
    #pragma unroll
    for (int k = 0; k < EF; k += 4) {
        v2f a = *(const v2f*)(aL + k);
        v2f b = *(const v2f*)(brow + k);
        acc = wmma_f32(a, b, acc);
    }
    const int m0 = half * 8;              // local edge row base for this half-wave
    #pragma unroll
    for (int r = 0; r < 8; ++r) {
        const int le = m0 + r;            // local edge 0..15
        const int s  = ssrc[le];
        const int d  = sdst[le];
        const float v = (feat_src[(size_t)s * HD + colg] + acc[r]) * sa[le][h];
        unsafeAtomicAdd(out + (size_t)d * HD + colg, v);
    }
}

// ---------------- launch ----------------
extern "C" void kernel_launch(void* const* d_in, const int* in_sizes, int n_in,
                              void* d_out, int out_size, void* d_ws, size_t ws_size,
                              hipStream_t stream) {
    const float* feat      = (const float*)d_in[0];
    const float* edge_feat = (const float*)d_in[1];
    const int*   src       = (const int*)d_in[2];
    const int*   dst       = (const int*)d_in[3];
    const float* Wfc       = (const float*)d_in[4];
    const float* Wedge     = (const float*)d_in[5];
    const float* attn_l    = (const float*)d_in[6];
    const float* attn_r    = (const float*)d_in[7];
    const float* attn_e    = (const float*)d_in[8];
    const float* Wres      = (const float*)d_in[9];
    const float* bias      = (const float*)d_in[10];
    float* out = (float*)d_out;

    float* ws       = (float*)d_ws;
    float* feat_src = ws + WS_FEATSRC;
    float* ebuf     = ws + WS_EBUF;
    float* el       = ws + WS_EL;
    float* er       = ws + WS_ER;
    float* emax     = ws + WS_EMAX;
    float* denom    = ws + WS_DENOM;
    float* wel      = ws + WS_WEL;
    float* wer      = ws + WS_WER;
    float* wee      = ws + WS_WEE;

    build_attn_w<<<1, 512, 0, stream>>>(Wfc, Wedge, attn_l, attn_r, attn_e,
                                        wel, wer, wee);
    init_redux<<<(NN * NH + 255) / 256, 256, 0, stream>>>(emax, denom);

    node_proj_wmma<<<dim3(NN / 16, 2), 256, 0, stream>>>(feat, Wfc, Wres, bias,
                                                         feat_src, out);
    node_attn<<<(NN * NH + 255) / 256, 256, 0, stream>>>(feat, wel, wer, el, er);

    edge_logits<<<(NE * NH + 255) / 256, 256, 0, stream>>>(edge_feat, src, dst,
                                                           wee, el, er, ebuf, emax);
    edge_softmax_num<<<(NE * NH + 255) / 256, 256, 0, stream>>>(dst, emax, ebuf,
                                                                denom);
    edge_agg_wmma<<<NE / 16, 256, 0, stream>>>(edge_feat, Wedge, src, dst,
                                               feat_src, ebuf, denom, out);
}